// GAE_NC_23536420782574
// MI455X (gfx1250) — compile-verified
//
#include <hip/hip_runtime.h>
#include <math.h>
#include <stdint.h>

typedef __attribute__((ext_vector_type(2))) float v2f;
typedef __attribute__((ext_vector_type(8))) float v8f;

#define IN_DIM  256
#define HID_DIM 256
#define OUT_DIM 128

// ---------------------------------------------------------------- CDNA5 async copy helpers
// GLOBAL_LOAD_ASYNC_TO_LDS_B128: per-lane copy of 16B global -> LDS, tracked by ASYNCcnt.
__device__ __forceinline__ void async_copy_b128(uint32_t lds_byte_off, const float* gptr) {
    asm volatile("global_load_async_to_lds_b128 %0, %1, off"
                 :: "v"(lds_byte_off), "v"(gptr)
                 : "memory");
}
__device__ __forceinline__ void wait_async0() {
    asm volatile("s_wait_asynccnt 0x0" ::: "memory");
}
// LDS aperture keeps the LDS byte offset in flat-address bits [31:0].
__device__ __forceinline__ uint32_t lds_off(const void* p) {
    return (uint32_t)(uintptr_t)p;
}

// ---------------------------------------------------------------- utilities
__global__ void k_zero(float* __restrict__ p, long long n) {
    long long i = (long long)blockIdx.x * blockDim.x + threadIdx.x;
    if (i < n) p[i] = 0.0f;
}

__global__ void k_deg(const int* __restrict__ src, const int* __restrict__ dst,
                      float* __restrict__ outdeg, float* __restrict__ indeg, int E) {
    int e = blockIdx.x * blockDim.x + threadIdx.x;
    if (e < E) {
        atomicAdd(&outdeg[src[e]], 1.0f);
        atomicAdd(&indeg[dst[e]], 1.0f);
    }
}

__global__ void k_norm(const float* __restrict__ outdeg, const float* __restrict__ indeg,
                       float* __restrict__ onorm, float* __restrict__ inorm, int n) {
    int i = blockIdx.x * blockDim.x + threadIdx.x;
    if (i < n) {
        onorm[i] = rsqrtf(fmaxf(outdeg[i], 1.0f));
        inorm[i] = rsqrtf(fmaxf(indeg[i], 1.0f));
    }
}

// ------------------------------------------------ WMMA fp32 GEMM (K == 256)
// C[M x Ncols] = (A * rowScale?) @ (transB ? B^T : B)
// A: M x K row-major. B: (!transB) K x Ncols row-major ; (transB) Ncols x K.
// 256 threads = 8 waves; block tile 64 rows x 128 cols; each wave a 32x32 tile.
// Double-buffered LDS staged with GLOBAL_LOAD_ASYNC_TO_LDS_B128 on the hot path.
#define APAD 36   // 36*4 = 144B row stride: 16B-aligned, walks all 16 banks for strided reads
#define BPAD 132  // 132*4 = 528B row stride: 16B-aligned

__global__ __launch_bounds__(256)
void k_gemm_wmma_f32(const float* __restrict__ A, const float* __restrict__ B,
                     float* __restrict__ C, const float* __restrict__ rowScale,
                     int M, int K, int Ncols, int transB) {
    __shared__ float As[2][64][APAD];
    __shared__ float Bs[2][32][BPAD];

    const int tid   = threadIdx.x;
    const int lane  = tid & 31;
    const int wave  = tid >> 5;
    const int l16   = lane & 15;
    const int hgrp  = lane >> 4;          // half-wave -> K-group select
    const int waveM = wave & 1;           // 32-row offset
    const int waveN = wave >> 1;          // 32-col offset
    const int rowBlock = blockIdx.x * 64;
    const int colBlock = blockIdx.y * 128;
    const bool fullTile = (rowBlock + 64 <= M);

    // loop-invariant per-lane row scales (out_norm), applied to A fragments post-LDS
    const int rowA0 = rowBlock + waveM * 32 + l16;
    const int rowA1 = rowA0 + 16;
    const float s0 = rowScale ? rowScale[rowA0 < M ? rowA0 : M - 1] : 1.0f;
    const float s1 = rowScale ? rowScale[rowA1 < M ? rowA1 : M - 1] : 1.0f;

    // stage one 32-wide K slab into LDS buffer `buf`
    auto stage = [&](int buf, int kb) {
        // ---- A tile: 64 x 32
        if (fullTile) {
            #pragma unroll
            for (int i = 0; i < 2; ++i) {
                int idx = tid + i * 256;
                int r = idx >> 3;             // 0..63
                int c = (idx & 7) * 4;        // 0..28
                async_copy_b128(lds_off(&As[buf][r][c]),
                                A + (long long)(rowBlock + r) * K + kb + c);
            }
        } else {
            #pragma unroll
            for (int i = 0; i < 2; ++i) {
                int idx = tid + i * 256;
                int r = idx >> 3;
                int c = (idx & 7) * 4;
                int gr = rowBlock + r;
                float4 v = make_float4(0.f, 0.f, 0.f, 0.f);
                if (gr < M) v = *(const float4*)(A + (long long)gr * K + kb + c);
                As[buf][r][c + 0] = v.x; As[buf][r][c + 1] = v.y;
                As[buf][r][c + 2] = v.z; As[buf][r][c + 3] = v.w;
            }
        }
        // ---- B tile: 32 x 128
        if (!transB) {
            #pragma unroll
            for (int i = 0; i < 4; ++i) {
                int idx = tid + i * 256;
                int r = idx >> 5;          // k: 0..31
                int c = (idx & 31) * 4;    // n: 0..124
                async_copy_b128(lds_off(&Bs[buf][r][c]),
                                B + (long long)(kb + r) * Ncols + colBlock + c);
            }
        } else {
            #pragma unroll
            for (int i = 0; i < 4; ++i) {
                int idx = tid + i * 256;
                int r = idx >> 3;          // n: 0..127 (row of fcW)
                int c = (idx & 7) * 4;     // k: 0..28
                float4 v = *(const float4*)(B + (long long)(colBlock + r) * K + kb + c);
                Bs[buf][c + 0][r] = v.x; Bs[buf][c + 1][r] = v.y;
                Bs[buf][c + 2][r] = v.z; Bs[buf][c + 3][r] = v.w;
            }
        }
    };

    v8f c00 = {}; v8f c01 = {}; v8f c10 = {}; v8f c11 = {};

    // prologue: fill buffer 0
    stage(0, 0);
    wait_async0();
    __syncthreads();

    int buf = 0;
    for (int kb = 0; kb < K; kb += 32) {
        // issue async copies for the next slab into the other buffer, then compute
        if (kb + 32 < K) stage(buf ^ 1, kb + 32);

        #pragma unroll
        for (int kk = 0; kk < 32; kk += 4) {
            const int ka = kk + 2 * hgrp;                     // K base for this lane half
            v2f a0, a1, b0, b1;
            a0[0] = As[buf][waveM * 32 + l16][ka];            // A frag: reg r holds K = 2*hgrp + r
            a0[1] = As[buf][waveM * 32 + l16][ka + 1];
            a1[0] = As[buf][waveM * 32 + 16 + l16][ka];
            a1[1] = As[buf][waveM * 32 + 16 + l16][ka + 1];
            a0[0] *= s0; a0[1] *= s0;                         // out_norm row scaling
            a1[0] *= s1; a1[1] *= s1;
            b0[0] = Bs[buf][ka][waveN * 32 + l16];            // B frag: reg r holds K = 2*hgrp + r
            b0[1] = Bs[buf][ka + 1][waveN * 32 + l16];
            b1[0] = Bs[buf][ka][waveN * 32 + 16 + l16];
            b1[1] = Bs[buf][ka + 1][waveN * 32 + 16 + l16];
            c00 = __builtin_amdgcn_wmma_f32_16x16x4_f32(false, a0, false, b0, (short)0, c00, false, false);
            c01 = __builtin_amdgcn_wmma_f32_16x16x4_f32(false, a0, false, b1, (short)0, c01, false, false);
            c10 = __builtin_amdgcn_wmma_f32_16x16x4_f32(false, a1, false, b0, (short)0, c10, false, false);
            c11 = __builtin_amdgcn_wmma_f32_16x16x4_f32(false, a1, false, b1, (short)0, c11, false, false);
        }

        wait_async0();        // next slab's copies (this wave's) are in LDS
        __syncthreads();      // everyone done computing on `buf` and done copying
        buf ^= 1;
    }

    // ---- store: C/D layout -> M = reg + 8*hgrp, N = l16
    const int baseRow = rowBlock + waveM * 32;
    const int baseCol = colBlock + waveN * 32 + l16;
    #pragma unroll
    for (int r = 0; r < 8; ++r) {
        int m0 = baseRow + r + 8 * hgrp;
        int m1 = m0 + 16;
        if (m0 < M) {
            C[(long long)m0 * Ncols + baseCol]      = c00[r];
            C[(long long)m0 * Ncols + baseCol + 16] = c01[r];
        }
        if (m1 < M) {
            C[(long long)m1 * Ncols + baseCol]      = c10[r];
            C[(long long)m1 * Ncols + baseCol + 16] = c11[r];
        }
    }
}

// --------------------------------------------------- SpMM: Agg[dst] += X[src]
__global__ void k_spmm_atomic(const float* __restrict__ X, const int* __restrict__ src,
                              const int* __restrict__ dst, float* __restrict__ Agg,
                              int E, int C) {
    const int cq = C >> 2;                                  // float4 groups per row
    long long t = (long long)blockIdx.x * blockDim.x + threadIdx.x;
    long long total = (long long)E * cq;
    if (t >= total) return;
    int e  = (int)(t / cq);
    int c4 = (int)(t - (long long)e * cq) << 2;
    int s = src[e], d = dst[e];
    float4 v = *(const float4*)(X + (long long)s * C + c4);
    float* o = Agg + (long long)d * C + c4;
    atomicAdd(o + 0, v.x);
    atomicAdd(o + 1, v.y);
    atomicAdd(o + 2, v.z);
    atomicAdd(o + 3, v.w);
}

// ------------------------------------------------------------- epilogues
__global__ void k_post1(const float* __restrict__ Agg, const float* __restrict__ inorm,
                        const float* __restrict__ b1, float* __restrict__ H,
                        int n, int C) {
    long long t = (long long)blockIdx.x * blockDim.x + threadIdx.x;
    if (t >= (long long)n * C) return;
    int row = (int)(t / C);
    int col = (int)(t - (long long)row * C);
    float v = Agg[t] * inorm[row] + b1[col];
    H[t] = v > 0.f ? v : 0.f;
}

__global__ void k_post2(const float* __restrict__ Agg2, const float* __restrict__ Agg3,
                        const float* __restrict__ inorm, const float* __restrict__ b2,
                        const float* __restrict__ b3, const float* __restrict__ noise,
                        float* __restrict__ Z, int n, int C) {
    long long t = (long long)blockIdx.x * blockDim.x + threadIdx.x;
    if (t >= (long long)n * C) return;
    int row = (int)(t / C);
    int col = (int)(t - (long long)row * C);
    float s    = inorm[row];
    float mean = fmaxf(Agg2[t] * s + b2[col], 0.f);
    float ls   = Agg3[t] * s + b3[col];
    Z[t] = mean + noise[t] * expf(ls);
}

// ---------------------------------------------------------------- driver
extern "C" void kernel_launch(void* const* d_in, const int* in_sizes, int n_in,
                              void* d_out, int out_size, void* d_ws, size_t ws_size,
                              hipStream_t stream) {
    const float* features = (const float*)d_in[0];
    const float* noise    = (const float*)d_in[1];
    const float* W1       = (const float*)d_in[2];
    const float* b1       = (const float*)d_in[3];
    const float* W2       = (const float*)d_in[4];
    const float* b2       = (const float*)d_in[5];
    const float* W3       = (const float*)d_in[6];
    const float* b3       = (const float*)d_in[7];
    const float* fcW      = (const float*)d_in[8];
    const int*   src      = (const int*)d_in[9];
    const int*   dst      = (const int*)d_in[10];

    const int N = in_sizes[0] / IN_DIM;     // 50000
    const int E = in_sizes[9];              // 800000

    // workspace layout (floats)
    float* ws     = (float*)d_ws;
    float* outdeg = ws;                                     // N
    float* indeg  = outdeg + N;                             // N
    float* onorm  = indeg + N;                              // N
    float* inorm  = onorm + N;                              // N
    float* buf1   = inorm + N;                              // N*256 (xw, then hw2|hw3)
    float* buf2   = buf1 + (long long)N * HID_DIM;          // N*256 (agg, then agg2|agg3)

    // output layout: z [N,128] | h [N,256] | seq_fts [N,256]
    float* z   = (float*)d_out;
    float* h   = z + (long long)N * OUT_DIM;
    float* seq = h + (long long)N * HID_DIM;

    const int T = 256;
    auto blocks = [](long long n, int t) { return (unsigned)((n + t - 1) / t); };

    // degrees + norms
    k_zero<<<blocks(2LL * N, T), T, 0, stream>>>(outdeg, 2LL * N);
    k_deg <<<blocks(E, T), T, 0, stream>>>(src, dst, outdeg, indeg, E);
    k_norm<<<blocks(N, T), T, 0, stream>>>(outdeg, indeg, onorm, inorm, N);

    dim3 gFull((N + 63) / 64, HID_DIM / 128);   // 256-col GEMMs
    dim3 gHalf((N + 63) / 64, OUT_DIM / 128);   // 128-col GEMMs

    // layer 1: xw = (features * onorm) @ W1 ; agg ; h = relu(agg*inorm + b1)
    k_gemm_wmma_f32<<<gFull, T, 0, stream>>>(features, W1, buf1, onorm, N, IN_DIM, HID_DIM, 0);
    k_zero<<<blocks((long long)N * HID_DIM, T), T, 0, stream>>>(buf2, (long long)N * HID_DIM);
    k_spmm_atomic<<<blocks((long long)E * (HID_DIM / 4), T), T, 0, stream>>>(buf1, src, dst, buf2, E, HID_DIM);
    k_post1<<<blocks((long long)N * HID_DIM, T), T, 0, stream>>>(buf2, inorm, b1, h, N, HID_DIM);

    // layers 2/3: hw2 = (h*onorm)@W2, hw3 = (h*onorm)@W3 ; agg ; z epilogue
    float* hw2  = buf1;
    float* hw3  = buf1 + (long long)N * OUT_DIM;
    float* agg2 = buf2;
    float* agg3 = buf2 + (long long)N * OUT_DIM;
    k_gemm_wmma_f32<<<gHalf, T, 0, stream>>>(h, W2, hw2, onorm, N, HID_DIM, OUT_DIM, 0);
    k_gemm_wmma_f32<<<gHalf, T, 0, stream>>>(h, W3, hw3, onorm, N, HID_DIM, OUT_DIM, 0);
    k_zero<<<blocks(2LL * N * OUT_DIM, T), T, 0, stream>>>(agg2, 2LL * N * OUT_DIM);
    k_spmm_atomic<<<blocks((long long)E * (OUT_DIM / 4), T), T, 0, stream>>>(hw2, src, dst, agg2, E, OUT_DIM);
    k_spmm_atomic<<<blocks((long long)E * (OUT_DIM / 4), T), T, 0, stream>>>(hw3, src, dst, agg3, E, OUT_DIM);
    k_post2<<<blocks((long long)N * OUT_DIM, T), T, 0, stream>>>(agg2, agg3, inorm, b2, b3, noise, z, N, OUT_DIM);

    // seq_fts = features @ fcW^T
    k_gemm_wmma_f32<<<gFull, T, 0, stream>>>(features, fcW, seq, nullptr, N, IN_DIM, IN_DIM, 1);
}